// AutoCorrelation_42142219108442
// MI455X (gfx1250) — compile-verified
//
#include <hip/hip_runtime.h>

typedef __attribute__((ext_vector_type(2))) float v2f;
typedef __attribute__((ext_vector_type(8))) float v8f;

#define BB    16
#define LL    2048
#define NH    8
#define DH    64
#define SSTR  (NH*DH)          /* 512 floats between consecutive t of a series */
#define NSER  (BB*NH*DH)       /* 8192 series */
#define TOPK  7                /* floor(log(2048)) */

// ---------------------------------------------------------------------------
// Kernel 1: circular cross-correlation via WMMA f32 16x16x4.
// One workgroup per (b,h,d) series. q,k staged doubled in LDS (no modulo in
// the hot loop). Wave w computes the 256 lags tau = w*256 + 16n + m as a
// single 16x16 GEMM with K=2048 (512 accumulating v_wmma_f32_16x16x4_f32).
// ---------------------------------------------------------------------------
__global__ __launch_bounds__(256) void corr_wmma_kernel(const float* __restrict__ q,
                                                        const float* __restrict__ k,
                                                        float* __restrict__ corr) {
    __shared__ float qs2[2 * LL];   // qs2[i] = q[i % L], i in [0,4096)
    __shared__ float ks2[2 * LL];   // ks2[i] = k[i % L]

    const int tid = threadIdx.x;
    const int sid = blockIdx.x;                 // series id
    const int b   = sid >> 9;                   // / (NH*DH)
    const int hd  = sid & (SSTR - 1);           // h*DH + d
    const long base = (long)b * ((long)LL * SSTR) + hd;
    const float* qg = q + base;
    const float* kg = k + base;

    // prefetch hint (global_prefetch_b8) spread across the series
    __builtin_prefetch(qg + (long)tid * SSTR * 8, 0, 0);
    __builtin_prefetch(kg + (long)tid * SSTR * 8, 0, 0);

    for (int i = tid; i < 2 * LL; i += 256) {
        const long t = (long)(i & (LL - 1)) * SSTR;
        qs2[i] = qg[t];
        ks2[i] = kg[t];
    }
    __syncthreads();

    const int lane = tid & 31;
    const int wave = tid >> 5;
    const int tau0 = wave * 256;
    const int m16  = lane & 15;     // A row m / B,C column n
    const int grp  = lane >> 4;     // lane half: K-pair (2g,2g+1) / C rows 8g..8g+7

    // A[m,s] = qs2[s + tau0 + m], lane holds K-pair starting at K=2*grp
    const int aidx = tau0 + m16 + grp * 2;
    // B[s,n] = ks2[s - 16n + 2048], lane n=m16 holds row pair kk = 2*grp,2*grp+1
    const int bidx = 2048 - 16 * m16 + grp * 2;

    v8f c = {};
    #pragma unroll 8
    for (int s = 0; s < LL; s += 4) {
        v2f a, bv;
        a.x  = qs2[aidx + s];
        a.y  = qs2[aidx + s + 1];
        bv.x = ks2[bidx + s];
        bv.y = ks2[bidx + s + 1];
        c = __builtin_amdgcn_wmma_f32_16x16x4_f32(false, a, false, bv,
                                                  (short)0, c, false, false);
    }

    // C layout: lane (grp,m16): n=m16, VGPR j holds m = 8*grp + j
    float* cb = corr + base + (long)tau0 * SSTR;
    #pragma unroll
    for (int j = 0; j < 8; ++j) {
        const int tau_rel = 16 * m16 + 8 * grp + j;
        cb[(long)tau_rel * SSTR] = c[j];
    }
}

// ---------------------------------------------------------------------------
// Kernel 2: per-series top-7 over corr, softmax, delay-gather aggregation.
// ---------------------------------------------------------------------------
__global__ __launch_bounds__(256) void topk_agg_kernel(const float* __restrict__ corr,
                                                       const float* __restrict__ v,
                                                       float* __restrict__ vnew) {
    __shared__ float cs[LL];
    __shared__ float wsh[TOPK];
    __shared__ int   dsh[TOPK];

    const int tid = threadIdx.x;
    const int sid = blockIdx.x;
    const int b   = sid >> 9;
    const int hd  = sid & (SSTR - 1);
    const long base = (long)b * ((long)LL * SSTR) + hd;

    const float* cg = corr + base;
    for (int i = tid; i < LL; i += 256) cs[i] = cg[(long)i * SSTR];
    __syncthreads();

    if (tid < 32) {
        const int lane = tid;
        for (int p = 0; p < TOPK; ++p) {
            float best = -3.4e38f;
            int   bi   = 0;
            for (int i = lane; i < LL; i += 32) {
                const float val = cs[i];
                if (val > best) { best = val; bi = i; }
            }
            #pragma unroll
            for (int off = 16; off > 0; off >>= 1) {
                const float ov = __shfl_down(best, off, 32);
                const int   oi = __shfl_down(bi,   off, 32);
                if (ov > best) { best = ov; bi = oi; }
            }
            bi = __shfl(bi, 0, 32);
            if (lane == 0) { wsh[p] = best; dsh[p] = bi; }
            cs[bi] = -3.4e38f;   // mask winner (uniform value, per-wave DS order)
        }
        if (lane == 0) {
            const float mx = wsh[0];
            float e[TOPK], sum = 0.f;
            #pragma unroll
            for (int p = 0; p < TOPK; ++p) { e[p] = __expf(wsh[p] - mx); sum += e[p]; }
            const float inv = 1.f / sum;
            #pragma unroll
            for (int p = 0; p < TOPK; ++p) wsh[p] = e[p] * inv;
        }
    }
    __syncthreads();

    float w[TOPK]; int dl[TOPK];
    #pragma unroll
    for (int p = 0; p < TOPK; ++p) { w[p] = wsh[p]; dl[p] = dsh[p]; }

    const float* vg = v    + base;
    float*       og = vnew + base;
    for (int t = tid; t < LL; t += 256) {
        float acc = 0.f;
        #pragma unroll
        for (int p = 0; p < TOPK; ++p) {
            acc += w[p] * vg[(long)((t + dl[p]) & (LL - 1)) * SSTR];
        }
        og[(long)t * SSTR] = acc;
    }
}

// ---------------------------------------------------------------------------
extern "C" void kernel_launch(void* const* d_in, const int* in_sizes, int n_in,
                              void* d_out, int out_size, void* d_ws, size_t ws_size,
                              hipStream_t stream) {
    const float* q = (const float*)d_in[0];
    const float* k = (const float*)d_in[1];
    const float* v = (const float*)d_in[2];

    float* vnew = (float*)d_out;                       // output 0: v_new
    float* corr = vnew + (size_t)BB * LL * NH * DH;    // output 1: corr

    corr_wmma_kernel<<<NSER, 256, 0, stream>>>(q, k, corr);
    topk_agg_kernel <<<NSER, 256, 0, stream>>>(corr, v, vnew);
}